// RIVEEncoder_43258910605316
// MI455X (gfx1250) — compile-verified
//
#include <hip/hip_runtime.h>
#include <math.h>

// MI455X notes:
//  - wave32 only: all shuffles use width 32, block = 256 threads = 8 waves.
//  - bandwidth-bound problem (201 MB in / 86 KB out, ~8.6 us floor @ 23.3 TB/s):
//    each input float is read exactly once via coalesced float4 (b128) loads.
//  - WMMA (V_WMMA_F32_16X16X4_F32) used for the Legendre least-squares
//    projection c = Minv[9x60] @ phi[60x8], padded to 16x64 @ 64x16.

typedef __attribute__((ext_vector_type(2))) float v2f;
typedef __attribute__((ext_vector_type(8))) float v8f;

static constexpr int kH = 256;
static constexpr int kW = 256;
static constexpr int kM = 60;   // number of crops
static constexpr int kP1 = 9;   // P+1 Legendre coefficients

// ---------------------------------------------------------------------------
// Setup kernel: build Minv = (L^T L)^-1 L^T  (static geometry), padded to
// [16][64] f32 row-major in d_ws. Exact double-precision Gauss-Jordan.
// ---------------------------------------------------------------------------
__global__ void rive_setup(float* __restrict__ ws) {
  if (blockIdx.x != 0 || threadIdx.x != 0) return;
  double L[kM][kP1];
  for (int t = 0; t < kM; ++t) {
    double x = 2.0 * ((double)t / (double)(kM - 1)) - 1.0;
    L[t][0] = 1.0;
    L[t][1] = x;
    for (int j = 1; j < kP1 - 1; ++j)
      L[t][j + 1] = ((2.0 * j + 1.0) * x * L[t][j] - (double)j * L[t][j - 1]) / (double)(j + 1);
  }
  double A[kP1][kP1];
  for (int i = 0; i < kP1; ++i)
    for (int j = 0; j < kP1; ++j) {
      double s = 0.0;
      for (int t = 0; t < kM; ++t) s += L[t][i] * L[t][j];
      A[i][j] = s;
    }
  // Gauss-Jordan inverse with partial pivoting
  double aug[kP1][2 * kP1];
  for (int i = 0; i < kP1; ++i) {
    for (int j = 0; j < kP1; ++j) aug[i][j] = A[i][j];
    for (int j = 0; j < kP1; ++j) aug[i][kP1 + j] = (i == j) ? 1.0 : 0.0;
  }
  for (int col = 0; col < kP1; ++col) {
    int piv = col;
    double mx = fabs(aug[col][col]);
    for (int r = col + 1; r < kP1; ++r) {
      double v = fabs(aug[r][col]);
      if (v > mx) { mx = v; piv = r; }
    }
    if (piv != col)
      for (int j = 0; j < 2 * kP1; ++j) {
        double tmp = aug[col][j]; aug[col][j] = aug[piv][j]; aug[piv][j] = tmp;
      }
    double d = aug[col][col];
    for (int j = 0; j < 2 * kP1; ++j) aug[col][j] /= d;
    for (int r = 0; r < kP1; ++r)
      if (r != col) {
        double f = aug[r][col];
        for (int j = 0; j < 2 * kP1; ++j) aug[r][j] -= f * aug[col][j];
      }
  }
  for (int i = 0; i < 16 * 64; ++i) ws[i] = 0.0f;
  for (int j = 0; j < kP1; ++j)
    for (int k = 0; k < kM; ++k) {
      double s = 0.0;
      for (int i = 0; i < kP1; ++i) s += aug[j][kP1 + i] * L[k][i];
      ws[j * 64 + k] = (float)s;
    }
}

// ---------------------------------------------------------------------------
// Main kernel: one block per sample. 8 waves, each wave owns 4 rows per
// 32-row chunk; per-row trimmed sums come from wave-level prefix scans whose
// values are stored in LDS only at 129 column positions (j in [0,64] U [193,256])
// covering every column the crop windows touch.
// ---------------------------------------------------------------------------
__global__ __launch_bounds__(256) void rive_main(const float* __restrict__ img,
                                                 const float* __restrict__ Aw,
                                                 float* __restrict__ out) {
  // prefix arrays: index i<=64 holds prefix(j=i), index i>=65 holds prefix(j=i+127)
  __shared__ float Pg[32][130];
  __shared__ float Pq[32][130];
  __shared__ float phiS[64][16];   // B matrix for WMMA: rows=K (crop k), cols=N (feature)
  __shared__ float accT[kM], accT2[kM], accL[kM], accR[kM], accTop[kM], accBot[kM];
  __shared__ float apmv[4][32];

  const int b = blockIdx.x;
  const int tid = threadIdx.x;
  const int lane = tid & 31;
  const int wv = tid >> 5;
  const float* base = img + (size_t)b * 3 * kH * kW;

  if (tid < kM) {
    accT[tid] = 0.f; accT2[tid] = 0.f; accL[tid] = 0.f;
    accR[tid] = 0.f; accTop[tid] = 0.f; accBot[tid] = 0.f;
  }
  __syncthreads();

  for (int c = 0; c < 8; ++c) {
    for (int rr = 0; rr < 4; ++rr) {
      const int rl = wv * 4 + rr;      // 0..31 within chunk
      const int R = c * 32 + rl;       // global row
      float carG = 0.f, carQ = 0.f;
      #pragma unroll
      for (int seg = 0; seg < 2; ++seg) {
        const int col = seg * 128 + lane * 4;
        const float4 r4 = *reinterpret_cast<const float4*>(base + (size_t)(0 * kH + R) * kW + col);
        const float4 g4 = *reinterpret_cast<const float4*>(base + (size_t)(1 * kH + R) * kW + col);
        const float4 b4 = *reinterpret_cast<const float4*>(base + (size_t)(2 * kH + R) * kW + col);
        float g0 = 0.299f * r4.x + 0.587f * g4.x + 0.114f * b4.x;
        float g1 = 0.299f * r4.y + 0.587f * g4.y + 0.114f * b4.y;
        float g2 = 0.299f * r4.z + 0.587f * g4.z + 0.114f * b4.z;
        float g3 = 0.299f * r4.w + 0.587f * g4.w + 0.114f * b4.w;
        float sg0 = g0, sg1 = sg0 + g1, sg2 = sg1 + g2, sg3 = sg2 + g3;
        float q0 = g0 * g0, q1 = g1 * g1, q2 = g2 * g2, q3 = g3 * g3;
        float sq0 = q0, sq1 = sq0 + q1, sq2 = sq1 + q2, sq3 = sq2 + q3;
        // wave32 inclusive scan of per-lane sums
        float incG = sg3, incQ = sq3;
        #pragma unroll
        for (int off = 1; off < 32; off <<= 1) {
          float tg = __shfl_up(incG, off, 32);
          float tq = __shfl_up(incQ, off, 32);
          if (lane >= off) { incG += tg; incQ += tq; }
        }
        const float baseG = carG + (incG - sg3);   // prefix before this lane's cols
        const float baseQ = carQ + (incQ - sq3);
        carG += __shfl(incG, 31, 32);
        carQ += __shfl(incQ, 31, 32);
        if (seg == 0) {
          if (lane < 16) {                         // j = 4*lane+1 .. 4*lane+4 (<=64)
            Pg[rl][4 * lane + 1] = baseG + sg0;  Pq[rl][4 * lane + 1] = baseQ + sq0;
            Pg[rl][4 * lane + 2] = baseG + sg1;  Pq[rl][4 * lane + 2] = baseQ + sq1;
            Pg[rl][4 * lane + 3] = baseG + sg2;  Pq[rl][4 * lane + 3] = baseQ + sq2;
            Pg[rl][4 * lane + 4] = baseG + sg3;  Pq[rl][4 * lane + 4] = baseQ + sq3;
          }
          if (lane == 0) { Pg[rl][0] = 0.f; Pq[rl][0] = 0.f; }
        } else {
          if (lane >= 16) {                        // j = 129+4*lane+i -> idx = j-127
            const int ix = 2 + 4 * lane;
            Pg[rl][ix + 0] = baseG + sg0;  Pq[rl][ix + 0] = baseQ + sq0;
            Pg[rl][ix + 1] = baseG + sg1;  Pq[rl][ix + 1] = baseQ + sq1;
            Pg[rl][ix + 2] = baseG + sg2;  Pq[rl][ix + 2] = baseQ + sq2;
            Pg[rl][ix + 3] = baseG + sg3;  Pq[rl][ix + 3] = baseQ + sq3;
          }
        }
      }
    }
    __syncthreads();
    if (tid < kM) {
      const int k = tid;
      float t_ = accT[k], t2_ = accT2[k], l_ = accL[k], r_ = accR[k];
      float tp = accTop[k], bt = accBot[k];
      for (int rl = 0; rl < 32; ++rl) {
        const int R = c * 32 + rl;
        if (R >= k && R <= 255 - k) {
          const float pa = Pg[rl][k];
          const float pb = Pg[rl][129 - k];        // prefix(j=256-k)
          const float inner = pb - pa;
          t_ += inner;
          t2_ += Pq[rl][129 - k] - Pq[rl][k];
          l_ += Pg[rl][k + 1] - pa;                // g[R,k]
          r_ += pb - Pg[rl][128 - k];              // g[R,255-k]
          if (R == k) tp = inner;
          if (R == 255 - k) bt = inner;
        }
      }
      accT[k] = t_; accT2[k] = t2_; accL[k] = l_; accR[k] = r_;
      accTop[k] = tp; accBot[k] = bt;
    }
    __syncthreads();
  }

  // ---- phi features (60x8), zero-padded to 64x16 for WMMA B operand -------
  if (tid < 64) {
    #pragma unroll
    for (int i = 0; i < 16; ++i) phiS[tid][i] = 0.f;
  }
  if (tid < kM) {
    const int k = tid;
    const float s = 256.f - 2.f * (float)k;
    const float cnt = s * s;
    const float tot = accT[k], tot2 = accT2[k];
    const float mu = tot / cnt;
    const float var = (tot2 - tot * tot / cnt) / (cnt - 1.f);
    const float sigma = sqrtf(fmaxf(var, 0.f));
    const float topm = accTop[k] / s, botm = accBot[k] / s;
    const float lefm = accL[k] / s,  rigm = accR[k] / s;
    const float flux = fabsf(topm - botm) + fabsf(lefm - rigm);
    const float area = cnt;
    const float z = 1.f - s / 256.f;
    const float theta = atan2f(s, fmaxf(256.f - s, 1.f));
    const float omz = 1.f - z;
    const float jac = area / fmaxf(omz * omz, 1e-4f);
    const float zc = flux / (area * fmaxf(1.f / s, 1e-4f) + 1e-6f);
    phiS[k][0] = mu;   phiS[k][1] = sigma; phiS[k][2] = area; phiS[k][3] = z;
    phiS[k][4] = theta; phiS[k][5] = flux; phiS[k][6] = jac;  phiS[k][7] = zc;
  }
  // ---- APM ray gather (4 diagonal rays x 32 samples) ----------------------
  if (tid < 128) {
    const int ray = tid >> 5, i = tid & 31;
    const float sxs[4] = {-1.f, 1.f, 1.f, -1.f};
    const float sys[4] = { 1.f, 1.f, -1.f, -1.f};
    const float f = (float)i / 31.f;
    const float xf = 128.f + sxs[ray] * f * 128.f;
    const float yf = 128.f + sys[ray] * f * 128.f;
    const int x = (int)fminf(fmaxf(xf, 0.f), 255.f);
    const int y = (int)fminf(fmaxf(yf, 0.f), 255.f);
    const size_t off = (size_t)y * kW + x;
    apmv[ray][i] = 0.299f * base[off] + 0.587f * base[(size_t)kH * kW + off]
                 + 0.114f * base[2 * (size_t)kH * kW + off];
  }
  __syncthreads();

  float* ob = out + (size_t)b * 84;

  // ---- WMMA: C[16x16] = Minv_pad[16x64] @ phi_pad[64x16] (chained K=4) ----
  if (tid < 32) {
    const int rown = lane & 15;           // A row M / B col N for this lane
    const int koff = (lane >> 4) * 2;     // lanes 0-15: K+{0,1}; 16-31: K+{2,3}
    v8f acc = {0.f, 0.f, 0.f, 0.f, 0.f, 0.f, 0.f, 0.f};
    #pragma unroll
    for (int kb = 0; kb < 64; kb += 4) {
      v2f a, bb;
      a.x = Aw[rown * 64 + kb + koff];
      a.y = Aw[rown * 64 + kb + koff + 1];
      bb.x = phiS[kb + koff][rown];
      bb.y = phiS[kb + koff + 1][rown];
      acc = __builtin_amdgcn_wmma_f32_16x16x4_f32(false, a, false, bb,
                                                  (short)0, acc, false, false);
    }
    // C/D layout: VGPR i holds M=i (lanes 0-15) and M=i+8 (lanes 16-31), N=lane%16.
    // Output is c.T flattened: out[ch*9 + j] = C[j][ch], ch=0..7, j=0..8.
    if (lane < 8) {
      #pragma unroll
      for (int j = 0; j < 8; ++j) ob[lane * 9 + j] = acc[j];
    } else if (lane >= 16 && lane < 24) {
      ob[(lane - 16) * 9 + 8] = acc[0];   // M=8 row
    }
  }

  // ---- APM features (threads 64..71, one feature each) --------------------
  if (tid >= 64 && tid < 72) {
    const int f = tid - 64;
    float res;
    if (f < 4) {
      const int r0 = (f >> 1);            // 0: |NE-SW|, 1: |NW-SE|
      float mean = 0.f;
      for (int i = 0; i < 32; ++i) mean += fabsf(apmv[r0][i] - apmv[r0 + 2][i]);
      mean *= (1.f / 32.f);
      if ((f & 1) == 0) {
        res = mean;
      } else {
        float v = 0.f;
        for (int i = 0; i < 32; ++i) {
          const float dd = fabsf(apmv[r0][i] - apmv[r0 + 2][i]) - mean;
          v += dd * dd;
        }
        res = sqrtf(v * (1.f / 32.f));
      }
    } else {
      const int ray = f - 4;
      float mean = 0.f;
      for (int i = 0; i < 32; ++i) mean += apmv[ray][i];
      mean *= (1.f / 32.f);
      float num = 0.f, var = 0.f;
      for (int i = 0; i < 32; ++i) {
        const float dv = apmv[ray][i] - mean;
        num += ((float)i - 15.5f) * dv;   // sum(xc^2) for 0..31 = 2728
        var += dv * dv;
      }
      const float stdv = sqrtf(var * (1.f / 32.f));
      res = (stdv < 1e-6f) ? 0.f : num / 2728.f;
    }
    ob[72 + f] = res;
  }

  // ---- quadrant n_eff: areas = s^2 exactly => slope = 2.0 (static) --------
  if (tid >= 80 && tid < 84) ob[tid] = 2.0f;
}

// ---------------------------------------------------------------------------
extern "C" void kernel_launch(void* const* d_in, const int* in_sizes, int n_in,
                              void* d_out, int out_size, void* d_ws, size_t ws_size,
                              hipStream_t stream) {
  const float* img = (const float*)d_in[0];
  float* out = (float*)d_out;
  float* ws = (float*)d_ws;                       // 16*64 f32 = 4 KB for Minv
  const int B = in_sizes[0] / (3 * kH * kW);
  rive_setup<<<1, 64, 0, stream>>>(ws);
  rive_main<<<B, 256, 0, stream>>>(img, ws, out);
}